// VertexingTask_6545530159481
// MI455X (gfx1250) — compile-verified
//
#include <hip/hip_runtime.h>
#include <math.h>

typedef float v2f __attribute__((ext_vector_type(2)));
typedef float v8f __attribute__((ext_vector_type(8)));

#define Bsz 8
#define Nn 128
#define Dd 128
#define Cc 32
#define Hh 256
#define PAIRS 16256              // 128*127
#define TOTPAIRS (Bsz * PAIRS)   // 130048
#define MROWS (Bsz * Nn)         // 1024
#define UVCOLS (2 * Hh)          // 512

// ---------------------------------------------------------------------------
// Phase 1: UV = X @ [W1a | W1b]   (M=1024, K=128, N=512) via v_wmma_f32_16x16x4_f32
// One wave per 16x16 output tile; 8 waves / block; 2048 tiles -> 256 blocks.
// A-frag layout (16x4 f32): lane holds row M=lane%16, K = {0,1}+2*(lane/16).
// B-frag layout (4x16 f32): lane holds col N=lane%16, K = {0,1}+2*(lane/16).
// D layout: VGPR r, lane -> M = r + 8*(lane/16), N = lane%16.
// ---------------------------------------------------------------------------
__global__ __launch_bounds__(256) void k_gemm_uv(const float* __restrict__ X,
                                                 const float* __restrict__ W1,
                                                 float* __restrict__ UV) {
  const int lane = threadIdx.x & 31;
  const int wave = threadIdx.x >> 5;
  const int tile = blockIdx.x * 8 + wave;   // 0..2047
  const int mt   = tile >> 5;               // 64 M-tiles
  const int nt   = tile & 31;               // 32 N-tiles (N=512)
  const int half = lane >> 4;               // 0 or 1
  const int l16  = lane & 15;

  const int arow = mt * 16 + l16;           // row of X for this lane's A frag
  const int ncol = nt * 16 + l16;           // output column 0..511
  // cols 0..255 -> U uses W1 rows 0..127; cols 256..511 -> V uses rows 128..255
  const float* bptr = (ncol < Hh) ? (W1 + ncol) : (W1 + Dd * Hh + (ncol - Hh));
  const float* aptr = X + arow * Dd + 2 * half;

  v8f acc = {};
  for (int k = 0; k < Dd; k += 4) {
    v2f a = *(const v2f*)(aptr + k);        // contiguous float2: K = k+2*half, +1
    v2f bv;
    bv.x = bptr[(k + 2 * half + 0) * Hh];
    bv.y = bptr[(k + 2 * half + 1) * Hh];
    acc = __builtin_amdgcn_wmma_f32_16x16x4_f32(false, a, false, bv,
                                                (short)0, acc, false, false);
  }

  float* out = UV + (mt * 16 + 8 * half) * UVCOLS + nt * 16 + l16;
#pragma unroll
  for (int r = 0; r < 8; ++r)
    out[r * UVCOLS] = acc[r];
}

// ---------------------------------------------------------------------------
// Phase 1b: wvec[b,h] = b1[h] + ctx[b,:] @ W1[256:288, h]; also zero accumulators.
// ---------------------------------------------------------------------------
__global__ __launch_bounds__(256) void k_wvec(const float* __restrict__ ctx,
                                              const float* __restrict__ W1,
                                              const float* __restrict__ b1,
                                              float* __restrict__ wvec,
                                              float* __restrict__ accum) {
  const int b = blockIdx.x;
  const int h = threadIdx.x;
  if (b == 0 && h < 2) accum[h] = 0.0f;     // stream-ordered before phase 2
  float s = b1[h];
  const float* w = W1 + 2 * Dd * Hh + h;    // rows 256..287
  const float* c = ctx + b * Cc;
#pragma unroll
  for (int k = 0; k < Cc; ++k)
    s = fmaf(c[k], w[k * Hh], s);
  wvec[b * Hh + h] = s;
}

// ---------------------------------------------------------------------------
// Phase 2: one wave per pair. pred = sum_h relu(U_i + V_j + wvec)*W2 + b2,
// plus BCE loss partials accumulated through LDS f32 atomics -> global atomic.
// ---------------------------------------------------------------------------
__global__ __launch_bounds__(256) void k_pairs(const float* __restrict__ UV,
                                               const float* __restrict__ wvec,
                                               const float* __restrict__ W2,
                                               const float* __restrict__ b2,
                                               const int* __restrict__ labels,
                                               const int* __restrict__ origin,
                                               float* __restrict__ pred,
                                               float* __restrict__ accum) {
  __shared__ float s_wv[Bsz * Hh];          // 8 KB
  __shared__ float s_w2[Hh];                // 1 KB
  __shared__ float s_acc[2];

  for (int t = threadIdx.x; t < Bsz * Hh; t += blockDim.x) s_wv[t] = wvec[t];
  if (threadIdx.x < Hh) s_w2[threadIdx.x] = W2[threadIdx.x];
  if (threadIdx.x < 2)  s_acc[threadIdx.x] = 0.0f;
  __syncthreads();

  const int lane   = threadIdx.x & 31;
  const int gwave  = (blockIdx.x * blockDim.x + threadIdx.x) >> 5;
  const int nwaves = (gridDim.x * blockDim.x) >> 5;
  const float b2v  = b2[0];

  for (int g = gwave; g < TOTPAIRS; g += nwaves) {
    const int b  = g / PAIRS;
    const int p  = g - b * PAIRS;
    const int i  = p / 127;
    const int jr = p - i * 127;
    const int j  = jr + (jr >= i ? 1 : 0);  // skip the diagonal

    const float* urow = UV + (b * Nn + i) * UVCOLS;        // U half
    const float* vrow = UV + (b * Nn + j) * UVCOLS + Hh;   // V half
    const float* wv   = s_wv + b * Hh;

    float acc = 0.0f;
#pragma unroll
    for (int t = 0; t < Hh / 32; ++t) {
      const int h = lane + 32 * t;
      float hv = urow[h] + vrow[h] + wv[h];
      hv = fmaxf(hv, 0.0f);
      acc = fmaf(hv, s_w2[h], acc);
    }
#pragma unroll
    for (int off = 16; off > 0; off >>= 1)
      acc += __shfl_xor(acc, off, 32);

    if (lane == 0) {
      const float z = acc + b2v;
      pred[g] = z;
      const int li = labels[b * Nn + i], lj = labels[b * Nn + j];
      const float match = (li == lj && li >= 0 && lj >= 0) ? 1.0f : 0.0f;
      const int oi = origin[b * Nn + i], oj = origin[b * Nn + j];
      const int wi = (oi == 3 || oi == 4 || oi == 5) ? 1 : ((oi == 1) ? -1 : 0);
      const int wj = (oj == 3 || oj == 4 || oj == 5) ? 1 : ((oj == 1) ? -1 : 0);
      const float wgt = (float)(1 + (wi & wj));   // two's-complement AND, as reference
      const float bce = fmaxf(z, 0.0f) - z * match + log1pf(expf(-fabsf(z)));
      unsafeAtomicAdd(&s_acc[0], bce * wgt);      // ds_add_f32
      unsafeAtomicAdd(&s_acc[1], match);
    }
  }
  __syncthreads();
  if (threadIdx.x == 0) unsafeAtomicAdd(&accum[0], s_acc[0]);
  if (threadIdx.x == 1) unsafeAtomicAdd(&accum[1], s_acc[1]);
}

// ---------------------------------------------------------------------------
// Phase 3: loss = numerator / match_count (task_weight == 1.0)
// ---------------------------------------------------------------------------
__global__ void k_finalize(const float* __restrict__ accum, float* __restrict__ out) {
  out[TOTPAIRS] = accum[0] / accum[1];
}

extern "C" void kernel_launch(void* const* d_in, const int* in_sizes, int n_in,
                              void* d_out, int out_size, void* d_ws, size_t ws_size,
                              hipStream_t stream) {
  (void)in_sizes; (void)n_in; (void)out_size; (void)ws_size;
  const float* x      = (const float*)d_in[0];   // (8,128,128)
  const float* ctx    = (const float*)d_in[1];   // (8,32)
  const float* W1     = (const float*)d_in[2];   // (288,256)
  const float* b1     = (const float*)d_in[3];   // (256,)
  const float* W2     = (const float*)d_in[4];   // (256,1)
  const float* b2     = (const float*)d_in[5];   // (1,)
  const int*   labels = (const int*)d_in[6];     // (8,128)
  const int*   origin = (const int*)d_in[7];     // (8,128)

  float* ws    = (float*)d_ws;
  float* UV    = ws;                             // 1024*512 = 524288 floats (2 MB)
  float* wvec  = ws + (size_t)MROWS * UVCOLS;    // 8*256 = 2048 floats
  float* accum = wvec + Bsz * Hh;                // 2 floats
  float* pred  = (float*)d_out;                  // 130048 preds + 1 loss

  k_gemm_uv <<<256, 256, 0, stream>>>(x, W1, UV);
  k_wvec    <<<Bsz, Hh, 0, stream>>>(ctx, W1, b1, wvec, accum);
  k_pairs   <<<512, 256, 0, stream>>>(UV, wvec, W2, b2, labels, origin, pred, accum);
  k_finalize<<<1, 1, 0, stream>>>(accum, pred);
}